// _VariationalQHead_65481071396152
// MI455X (gfx1250) — compile-verified
//
#include <hip/hip_runtime.h>

// CDNA5 wave32 WMMA types
typedef __attribute__((ext_vector_type(2))) float v2f;
typedef __attribute__((ext_vector_type(8))) float v8f;

// 4-qubit variational circuit head, reduced to closed form:
//   All post-RY gates are diagonal (RZ) or basis permutations (CNOT), so
//   |state|^2 is independent of `params`. out = sigma(4x16) * probs(16x16batch),
//   probs(t) = prod_k w_k[t_k], w_k = (1 +/- cos x_k)/2,
//   sigma[i][t] = (-1)^popc(t & mask_i), masks = {0x7,0xC,0xE,0xF}
//   (t = b0*8+b1*4+b2*2+b3, derived from CNOT(0,1),(1,2),(2,3),(3,0) bit flow).
// The 16-state -> 4-channel signed reduction is done exactly in f32 with
// V_WMMA_F32_16X16X4_F32, K-chained over 4 groups of 4 states.
__global__ __launch_bounds__(256) void vqhead_wmma_kernel(
    const float* __restrict__ x, float* __restrict__ out,
    int nTiles, int nBatch) {
  const int lane = threadIdx.x & 31;
  const int wave = threadIdx.x >> 5;
  const int tile = blockIdx.x * 8 + wave;   // 8 waves / block, 16 batch / wave
  if (tile >= nTiles) return;               // wave-uniform: EXEC stays all-1s

  const int half = lane >> 4;               // K half-wave split (states 4j+{0,1} vs {2,3})
  const int col  = lane & 15;               // A: channel row (M); B: batch column (N)
  int b = tile * 16 + col;
  if (b >= nBatch) b = nBatch - 1;          // clamp (B = 1M is a multiple of 16 anyway)

  // x[b][0..3] -- both half-waves read the same 16 float4s (same cache lines).
  const float4 xv = ((const float4*)x)[b];
  const float c0 = __cosf(xv.x), c1 = __cosf(xv.y);
  const float c2 = __cosf(xv.z), c3 = __cosf(xv.w);
  // w_k[bit]: probability qubit k collapses to `bit` after RY(x_k)
  const float w0a = 0.5f * (1.f + c0), w0b = 0.5f * (1.f - c0);
  const float w1a = 0.5f * (1.f + c1), w1b = 0.5f * (1.f - c1);
  const float w2a = 0.5f * (1.f + c2), w2b = 0.5f * (1.f - c2);
  const float w3a = 0.5f * (1.f + c3), w3b = 0.5f * (1.f - c3);

  // p01[j] = w0[(j>>1)&1] * w1[j&1]  (b0,b1 are the high bits of t = 4j+...)
  float p01[4];
  p01[0] = w0a * w1a; p01[1] = w0a * w1b;
  p01[2] = w0b * w1a; p01[3] = w0b * w1b;
  // (b2,b3) = (half, r) for this lane's two K slots
  const float w2h = half ? w2b : w2a;
  const float q0 = w2h * w3a;               // r = 0 -> b3 = 0
  const float q1 = w2h * w3b;               // r = 1 -> b3 = 1

  // Channel parity masks over t's bits (qubit0 = bit3 ... qubit3 = bit0)
  const int masks[4] = {0x7, 0xC, 0xE, 0xF};
  const int   m     = masks[col & 3];
  const float valid = (col < 4) ? 1.f : 0.f;  // zero A rows 4..15

  v8f acc = {0.f, 0.f, 0.f, 0.f, 0.f, 0.f, 0.f, 0.f};
#pragma unroll
  for (int j = 0; j < 4; ++j) {
    const int t0 = 4 * j + 2 * half;        // state index for VGPR r=0 (r=1 -> t0+1)
    v2f A, B;
    A.x = valid * ((__popc((unsigned)(t0 & m)) & 1) ? -1.f : 1.f);
    A.y = valid * ((__popc((unsigned)((t0 + 1) & m)) & 1) ? -1.f : 1.f);
    B.x = p01[j] * q0;                      // probs[t0][batch col]
    B.y = p01[j] * q1;                      // probs[t0+1][batch col]
    // D(16x16,f32) += A(16x4,f32) x B(4x16,f32); exact RNE f32 math
    acc = __builtin_amdgcn_wmma_f32_16x16x4_f32(
        false, A, false, B, (short)0, acc, false, false);
  }

  // D layout: VGPR r, lanes 0-15 hold M=r (channel r), N=lane (batch).
  if (half == 0 && (tile * 16 + col) < nBatch) {
    float4 o;
    o.x = acc[0]; o.y = acc[1]; o.z = acc[2]; o.w = acc[3];
    ((float4*)out)[b] = o;                  // coalesced 16B per batch element
  }
}

extern "C" void kernel_launch(void* const* d_in, const int* in_sizes, int n_in,
                              void* d_out, int out_size, void* d_ws, size_t ws_size,
                              hipStream_t stream) {
  const float* x = (const float*)d_in[0];
  // d_in[1] (params) provably cannot affect the output: RZ gates are diagonal
  // and CNOTs permute basis states, so measured probabilities are phase-free.
  float* out = (float*)d_out;
  const int nBatch = in_sizes[0] / 4;           // 1,048,576
  const int nTiles = (nBatch + 15) / 16;        // 16 batch elements per wave
  dim3 block(256);                              // 8 wave32 waves
  dim3 grid((nTiles + 7) / 8);
  vqhead_wmma_kernel<<<grid, block, 0, stream>>>(x, out, nTiles, nBatch);
}